// NetModular_85993835200734
// MI455X (gfx1250) — compile-verified
//
#include <hip/hip_runtime.h>
#include <hip/hip_bf16.h>
#include <math.h>
#include <stdint.h>

// Problem constants (match reference)
#define GG    10000      // graphs
#define N0_   32         // nodes per graph
#define E0_   64         // edges per graph
#define NF_   64         // input features
#define H_    128        // hidden
#define D_    256        // ddi hidden
#define B_    4096       // pos edges (neg follow)
#define TWOB  (2 * B_)
#define NTOT  (GG * N0_) // 320000
#define ETOT  (GG * E0_) // 640000

typedef float v2f __attribute__((ext_vector_type(2)));
typedef float v8f __attribute__((ext_vector_type(8)));

static inline int cdiv(int a, int b) { return (a + b - 1) / b; }

// ---------------------------------------------------------------------------
// f32 WMMA GEMM:  C[M,N] = act(A[M,K] @ W[K,N] + bias)
// One wave computes a 16x128 strip: 8 accumulator tiles, so each A fragment
// is loaded once per k-step and reused by 8 consecutive V_WMMA_F32_16X16X4_F32
// ops (8x less A traffic than tile-per-wave; A is the dominant HBM stream).
// act: 0 = none, 1 = relu, 2 = sigmoid. bias may be nullptr.
// M multiple of 16; N multiple of 128; K multiple of 4.
// ---------------------------------------------------------------------------
__global__ __launch_bounds__(256)
void gemm_f32_wmma(const float* __restrict__ A, const float* __restrict__ W,
                   const float* __restrict__ bias, float* __restrict__ C,
                   int M, int K, int N, int act)
{
    int lane = threadIdx.x & 31;
    int waveGlobal = blockIdx.x * 8 + (threadIdx.x >> 5);
    int stripsN = N >> 7;                       // strips of 128 columns
    int mTile = waveGlobal / stripsN;
    int nStrip = waveGlobal - mTile * stripsN;
    if (mTile * 16 >= M) return;                // wave-uniform -> EXEC all ones

    int m0 = mTile << 4;
    int n0 = nStrip << 7;
    int mrow = m0 + (lane & 15);
    int ncol = lane & 15;
    int kb = (lane >> 4) * 2;                   // lanes 0-15: K+0/1 ; 16-31: K+2/3

    v8f zero = {0.f,0.f,0.f,0.f,0.f,0.f,0.f,0.f};
    v8f acc[8];
#pragma unroll
    for (int t = 0; t < 8; t++) acc[t] = zero;

    const float* Arow = A + (size_t)mrow * K + kb;
    for (int k = 0; k < K; k += 4) {
        v2f a;
        a.x = Arow[k];
        a.y = Arow[k + 1];
        const float* Wr0 = W + (size_t)(k + kb) * N + n0 + ncol;
        const float* Wr1 = Wr0 + N;
#pragma unroll
        for (int t = 0; t < 8; t++) {
            v2f b;
            b.x = Wr0[t * 16];
            b.y = Wr1[t * 16];
            acc[t] = __builtin_amdgcn_wmma_f32_16x16x4_f32(
                         false, a, false, b, (short)0, acc[t], false, false);
        }
    }

    int half = lane >> 4;                       // rows +0 or +8
#pragma unroll
    for (int t = 0; t < 8; t++) {
        int col = n0 + t * 16 + ncol;
        float bval = bias ? bias[col] : 0.f;
#pragma unroll
        for (int v = 0; v < 8; v++) {
            int row = m0 + v + half * 8;
            float xv = acc[t][v] + bval;
            if (act == 1)      xv = fmaxf(xv, 0.f);
            else if (act == 2) xv = 1.f / (1.f + expf(-xv));
            C[(size_t)row * N + col] = xv;
        }
    }
}

// ---------------------------------------------------------------------------
// Small utility kernels
// ---------------------------------------------------------------------------
__global__ void fill_f32(float* p, float v, int n) {
    int i = blockIdx.x * 256 + threadIdx.x; if (i < n) p[i] = v;
}
__global__ void fill_i32(int* p, int v, int n) {
    int i = blockIdx.x * 256 + threadIdx.x; if (i < n) p[i] = v;
}

__global__ void deg_scatter(const int* __restrict__ dst, const float* __restrict__ ew,
                            float* __restrict__ deg, int E) {
    int e = blockIdx.x * 256 + threadIdx.x;
    if (e < E) atomicAdd(&deg[dst[e]], ew[e]);
}
__global__ void finish_dis(float* d, int n) {
    int i = blockIdx.x * 256 + threadIdx.x;
    if (i < n) d[i] = rsqrtf(d[i] + 1.0f);
}

// ---------------------------------------------------------------------------
// Per-graph GCN aggregation (block per graph). The h_lin tile (NPG x 128,
// contiguous) is staged into LDS with GLOBAL_LOAD_ASYNC_TO_LDS_B128
// (ASYNCcnt path), overlapped with edge-coefficient computation, then
// accumulated with LDS float atomics.
// out = relu( sum_e dis[s]*ew*dis[d]*hlin[s] + dis[n]^2*hlin[n] + b )
// ---------------------------------------------------------------------------
__global__ __launch_bounds__(256)
void gcn_agg_graph(const float* __restrict__ hlin, const float* __restrict__ dis,
                   const int* __restrict__ src, const int* __restrict__ dst,
                   const float* __restrict__ ew, const float* __restrict__ bias,
                   float* __restrict__ out, int NPG)
{
    __shared__ __align__(16) float sh[32 * 128];
    __shared__ float sacc[32 * 128];
    __shared__ float sdis[32];
    __shared__ float scoef[64];
    __shared__ int   ssrc[64], sdst[64];

    int g = blockIdx.x;
    int base = g * NPG;
    int tid = threadIdx.x;
    int nf = NPG * 128;
    int nbytes = nf * 4;                        // 4096 * NPG/8  (multiple of 4096)

    // async bulk copy: 16 B per lane per issue; global tile base is 512 B aligned
    const char* gsrc = (const char*)(hlin + (size_t)base * 128);
    char* lbase = (char*)sh;
    for (int bo = tid * 16; bo < nbytes; bo += 256 * 16) {
        unsigned ldsAddr = (unsigned)(uintptr_t)(lbase + bo);
        const void* gp = (const void*)(gsrc + bo);
        asm volatile("global_load_async_to_lds_b128 %0, %1, off"
                     :: "v"(ldsAddr), "v"(gp) : "memory");
    }

    if (tid < NPG) sdis[tid] = dis[base + tid];
    if (tid < 64) {
        int e = g * 64 + tid;
        int s = src[e], d = dst[e];
        ssrc[tid] = s - base;
        sdst[tid] = d - base;
        scoef[tid] = dis[s] * ew[e] * dis[d];
    }

    asm volatile("s_wait_asynccnt 0" ::: "memory");
    __syncthreads();

    for (int i = tid; i < nf; i += 256) {
        float d = sdis[i >> 7];
        sacc[i] = d * d * sh[i];
    }
    __syncthreads();

    for (int i = tid; i < 64 * 128; i += 256) {
        int e = i >> 7, f = i & 127;
        float c = scoef[e];
        if (c != 0.f)
            atomicAdd(&sacc[sdst[e] * 128 + f], c * sh[ssrc[e] * 128 + f]);
    }
    __syncthreads();

    for (int i = tid; i < nf; i += 256) {
        float v = sacc[i] + bias[i & 127];
        out[(size_t)base * 128 + i] = fmaxf(v, 0.f);
    }
}

// ---------------------------------------------------------------------------
// SAGPool scoring:  score = tanh( scatter_dst(ew * (h[src].Wn)) + bn + h.Wr )
// ---------------------------------------------------------------------------
__global__ __launch_bounds__(256)
void score_dot(const float* __restrict__ h, const float* __restrict__ Wn,
               const float* __restrict__ Wr, float* __restrict__ p,
               float* __restrict__ r, int N)
{
    int warp = (blockIdx.x * 256 + threadIdx.x) >> 5;
    int lane = threadIdx.x & 31;
    if (warp >= N) return;
    const float* row = h + (size_t)warp * 128;
    float pn = 0.f, pr = 0.f;
    for (int f = lane; f < 128; f += 32) {
        float x = row[f];
        pn += x * Wn[f];
        pr += x * Wr[f];
    }
    for (int o = 16; o; o >>= 1) { pn += __shfl_xor(pn, o); pr += __shfl_xor(pr, o); }
    if (lane == 0) { p[warp] = pn; r[warp] = pr; }
}

__global__ void score_scatter(const int* __restrict__ src, const int* __restrict__ dst,
                              const float* __restrict__ ew, const float* __restrict__ p,
                              float* __restrict__ sacc, int E)
{
    int e = blockIdx.x * 256 + threadIdx.x;
    if (e < E) atomicAdd(&sacc[dst[e]], ew[e] * p[src[e]]);
}

__global__ void score_finish(float* __restrict__ sacc, const float* __restrict__ r,
                             const float* __restrict__ bn, int N)
{
    int i = blockIdx.x * 256 + threadIdx.x;
    if (i < N) sacc[i] = tanhf(sacc[i] + bn[0] + r[i]);
}

// One thread per graph: top-k by score, descending, lowest-index tie-break
// (matches jax.lax.top_k). old2new pre-filled with -1.
__global__ void topk_kernel(const float* __restrict__ score, float* __restrict__ topv,
                            int* __restrict__ perm, int* __restrict__ old2new,
                            int G, int NPG, int k)
{
    int g = blockIdx.x * 256 + threadIdx.x;
    if (g >= G) return;
    float s[32]; int id[32];
    for (int i = 0; i < NPG; i++) { s[i] = score[g * NPG + i]; id[i] = i; }
    for (int i = 0; i < k; i++) {
        int best = i;
        for (int j = i + 1; j < NPG; j++) if (s[j] > s[best]) best = j;
        float ts = s[best]; s[best] = s[i]; s[i] = ts;
        int   ti = id[best]; id[best] = id[i]; id[i] = ti;
        topv[g * k + i] = s[i];
        perm[g * k + i] = g * NPG + id[i];
        old2new[g * NPG + id[i]] = g * k + i;
    }
}

// xk[i] = h[perm[i]] * topv[i]
__global__ void gather_gate(const float* __restrict__ h, const int* __restrict__ perm,
                            const float* __restrict__ topv, float* __restrict__ out,
                            int NK)
{
    int i = blockIdx.x * 256 + threadIdx.x;
    if (i >= NK * 128) return;
    int n = i >> 7, f = i & 127;
    out[i] = h[(size_t)perm[n] * 128 + f] * topv[n];
}

// remap edges through old2new; invalid edges -> first node of graph, ew = 0
// (same math as reference's index-0/weight-0; preserves graph locality).
__global__ void edge_remap(const int* __restrict__ osrc, const int* __restrict__ odst,
                           const float* __restrict__ oew, int* __restrict__ nsrc,
                           int* __restrict__ ndst, float* __restrict__ nw,
                           const int* __restrict__ old2new, int k, int E)
{
    int e = blockIdx.x * 256 + threadIdx.x;
    if (e >= E) return;
    int g = e >> 6;                                  // 64 edges per graph
    int s = old2new[osrc[e]];
    int d = old2new[odst[e]];
    float w = oew[e];
    bool valid = (s >= 0) && (d >= 0);
    nsrc[e] = valid ? s : g * k;
    ndst[e] = valid ? d : g * k;
    nw[e]   = valid ? w : 0.f;
}

// readout: feat[g, colOff+f] = max_j xk[g*k+j, f]; feat[g, colOff+128+f] = mean
__global__ void readout_kernel(const float* __restrict__ xk, float* __restrict__ feat,
                               int k, int colOff, int G)
{
    int i = blockIdx.x * 256 + threadIdx.x;
    if (i >= G * 128) return;
    int g = i >> 7, f = i & 127;
    const float* base = xk + (size_t)g * k * 128 + f;
    float mx = -INFINITY, sm = 0.f;
    for (int j = 0; j < k; j++) {
        float v = base[(size_t)j * 128];
        mx = fmaxf(mx, v); sm += v;
    }
    feat[(size_t)g * 768 + colOff + f]       = mx;
    feat[(size_t)g * 768 + colOff + 128 + f] = sm / (float)k;
}

// ---------------------------------------------------------------------------
// DDI GCN (unit edge weights, 256-dim)
// ---------------------------------------------------------------------------
__global__ void ddi_deg(const int* __restrict__ dd, float* __restrict__ deg, int E2) {
    int e = blockIdx.x * 256 + threadIdx.x;
    if (e < E2) atomicAdd(&deg[dd[e]], 1.0f);
}
__global__ void ddi_init(const float* __restrict__ hd, const float* __restrict__ dis,
                         float* __restrict__ zagg, int total) {
    int i = blockIdx.x * 256 + threadIdx.x;
    if (i >= total) return;
    float d = dis[i >> 8];
    zagg[i] = d * d * hd[i];
}
__global__ void ddi_scatter(const int* __restrict__ ds, const int* __restrict__ dd,
                            const float* __restrict__ dis, const float* __restrict__ hd,
                            float* __restrict__ zagg, int E2) {
    int i = blockIdx.x * 256 + threadIdx.x;
    if (i >= E2 * 256) return;
    int e = i >> 8, f = i & 255;
    float c = dis[ds[e]] * dis[dd[e]];
    atomicAdd(&zagg[(size_t)dd[e] * 256 + f], c * hd[(size_t)ds[e] * 256 + f]);
}
__global__ void bias_relu_256(const float* __restrict__ zagg, const float* __restrict__ bd,
                              float* __restrict__ z, int total) {
    int i = blockIdx.x * 256 + threadIdx.x;
    if (i < total) z[i] = fmaxf(zagg[i] + bd[i & 255], 0.f);
}
__global__ void gather_rows256(const float* __restrict__ z, const int* __restrict__ idx,
                               float* __restrict__ out, int rows) {
    int i = blockIdx.x * 256 + threadIdx.x;
    if (i >= rows * 256) return;
    int r = i >> 8, f = i & 255;
    out[i] = z[(size_t)idx[r] * 256 + f];
}

// ---------------------------------------------------------------------------
// Loss: l = X + A - Y per row; norms; pos_x copy. Warp per row.
// out layout: loss[B] | norm_pos[B] | norm_neg[B] | pos_x[B*256]
// ---------------------------------------------------------------------------
__global__ __launch_bounds__(256)
void loss_norm(const float* __restrict__ X, const float* __restrict__ Y,
               const float* __restrict__ Aa, float* __restrict__ out, int Bn)
{
    int warp = (blockIdx.x * 256 + threadIdx.x) >> 5;
    int lane = threadIdx.x & 31;
    if (warp >= 2 * Bn) return;
    const float* xr = X  + (size_t)warp * 256;
    const float* yr = Y  + (size_t)warp * 256;
    const float* ar = Aa + (size_t)warp * 256;
    float s = 0.f;
    for (int f = lane; f < 256; f += 32) {
        float l = xr[f] + ar[f] - yr[f];
        s += l * l;
        if (warp < Bn) out[(size_t)3 * Bn + (size_t)warp * 256 + f] = xr[f];
    }
    for (int o = 16; o; o >>= 1) s += __shfl_xor(s, o);
    if (lane == 0) {
        float nrm = sqrtf(s);
        if (warp < Bn) out[Bn + warp] = nrm;
        else           out[2 * Bn + (warp - Bn)] = nrm;
    }
}
__global__ void loss_final(float* __restrict__ out, int Bn) {
    int i = blockIdx.x * 256 + threadIdx.x;
    if (i < Bn) out[i] = 2.0f * (float)D_ - out[Bn + i] + 0.5f * out[2 * Bn + i];
}

// ---------------------------------------------------------------------------
extern "C" void kernel_launch(void* const* d_in, const int* in_sizes, int n_in,
                              void* d_out, int out_size, void* d_ws, size_t ws_size,
                              hipStream_t stream)
{
    (void)in_sizes; (void)n_in; (void)out_size; (void)ws_size;

    // --- inputs ---
    const float* x    = (const float*)d_in[0];
    const int*   esrc = (const int*)  d_in[1];
    const int*   edst = (const int*)  d_in[2];
    const float* ew0  = (const float*)d_in[3];
    const int*   ddi  = (const int*)  d_in[4];
    const float* attr = (const float*)d_in[5];
    const float* W1   = (const float*)d_in[6];   const float* b1  = (const float*)d_in[7];
    const float* Ws1r = (const float*)d_in[8];   const float* Ws1n= (const float*)d_in[9];
    const float* bs1  = (const float*)d_in[10];
    const float* W2   = (const float*)d_in[11];  const float* b2  = (const float*)d_in[12];
    const float* Ws2r = (const float*)d_in[13];  const float* Ws2n= (const float*)d_in[14];
    const float* bs2  = (const float*)d_in[15];
    const float* W3   = (const float*)d_in[16];  const float* b3  = (const float*)d_in[17];
    const float* Ws3r = (const float*)d_in[18];  const float* Ws3n= (const float*)d_in[19];
    const float* bs3  = (const float*)d_in[20];
    const float* Wd   = (const float*)d_in[21];  const float* bd  = (const float*)d_in[22];
    const float* Wl1  = (const float*)d_in[23];  const float* bl1 = (const float*)d_in[24];
    const float* Wl2  = (const float*)d_in[25];  const float* bl2 = (const float*)d_in[26];
    const float* Wl3  = (const float*)d_in[27];  const float* bl3 = (const float*)d_in[28];

    const int N1 = NTOT, N2 = GG * 16, N3 = GG * 8, N4 = GG * 4, E = ETOT;
    const int* ds = ddi;           // row 0 of [2, 2B]
    const int* dd = ddi + TWOB;    // row 1

    // --- workspace arena ---
    size_t off = 0;
    auto alloc = [&](size_t bytes) -> void* {
        void* p = (char*)d_ws + off;
        off += (bytes + 255) & ~(size_t)255;
        return p;
    };
    float* bigA = (float*)alloc((size_t)N1 * 128 * 4);   // 164 MB
    float* bigB = (float*)alloc((size_t)N1 * 128 * 4);   // 164 MB
    float* dis  = (float*)alloc((size_t)N1 * 4);
    float* pbuf = (float*)alloc((size_t)N1 * 4);
    float* rbuf = (float*)alloc((size_t)N1 * 4);
    float* sbuf = (float*)alloc((size_t)N1 * 4);
    int*   o2n  = (int*)  alloc((size_t)N1 * 4);
    int*   es   = (int*)  alloc((size_t)E * 4);
    int*   ed   = (int*)  alloc((size_t)E * 4);
    float* eew  = (float*)alloc((size_t)E * 4);
    int*   perm = (int*)  alloc((size_t)N2 * 4);
    float* topv = (float*)alloc((size_t)N2 * 4);
    float* feat = (float*)alloc((size_t)GG * 768 * 4);
    float* degd = (float*)alloc((size_t)GG * 4);
    // late-stage buffers alias bigA/bigB (dead after readout x3)
    float* hd   = bigA;
    float* zagg = hd   + (size_t)GG * 256;
    float* z    = zagg + (size_t)GG * 256;
    float* srcf = z    + (size_t)GG * 256;
    float* tgtf = srcf + (size_t)TWOB * 256;
    float* Xh   = bigB;
    float* Yh   = Xh + (size_t)TWOB * 256;
    float* Ah   = Yh + (size_t)TWOB * 256;

    float* out = (float*)d_out;

    auto gemm = [&](const float* A, const float* W, const float* bias, float* C,
                    int M, int K, int N, int act) {
        int waves = (M / 16) * (N >> 7);          // one wave per 16x128 strip
        gemm_f32_wmma<<<cdiv(waves, 8), 256, 0, stream>>>(A, W, bias, C, M, K, N, act);
    };

    // ================= Layer 1: GCN + SAGPool + readout =================
    gemm(x, W1, nullptr, bigA, N1, 64, 128, 0);                       // h_lin1
    fill_f32<<<cdiv(N1, 256), 256, 0, stream>>>(dis, 0.f, N1);
    deg_scatter<<<cdiv(E, 256), 256, 0, stream>>>(edst, ew0, dis, E);
    finish_dis<<<cdiv(N1, 256), 256, 0, stream>>>(dis, N1);
    gcn_agg_graph<<<GG, 256, 0, stream>>>(bigA, dis, esrc, edst, ew0, b1, bigB, 32); // h1

    score_dot<<<cdiv(N1 * 32, 256), 256, 0, stream>>>(bigB, Ws1n, Ws1r, pbuf, rbuf, N1);
    fill_f32<<<cdiv(N1, 256), 256, 0, stream>>>(sbuf, 0.f, N1);
    score_scatter<<<cdiv(E, 256), 256, 0, stream>>>(esrc, edst, ew0, pbuf, sbuf, E);
    score_finish<<<cdiv(N1, 256), 256, 0, stream>>>(sbuf, rbuf, bs1, N1);
    fill_i32<<<cdiv(N1, 256), 256, 0, stream>>>(o2n, -1, N1);
    topk_kernel<<<cdiv(GG, 256), 256, 0, stream>>>(sbuf, topv, perm, o2n, GG, 32, 16);
    gather_gate<<<cdiv(N2 * 128, 256), 256, 0, stream>>>(bigB, perm, topv, bigA, N2); // xk1
    readout_kernel<<<cdiv(GG * 128, 256), 256, 0, stream>>>(bigA, feat, 16, 0, GG);   // x1
    edge_remap<<<cdiv(E, 256), 256, 0, stream>>>(esrc, edst, ew0, es, ed, eew, o2n, 16, E);

    // ================= Layer 2 =================
    gemm(bigA, W2, nullptr, bigB, N2, 128, 128, 0);                   // h_lin2
    fill_f32<<<cdiv(N2, 256), 256, 0, stream>>>(dis, 0.f, N2);
    deg_scatter<<<cdiv(E, 256), 256, 0, stream>>>(ed, eew, dis, E);
    finish_dis<<<cdiv(N2, 256), 256, 0, stream>>>(dis, N2);
    gcn_agg_graph<<<GG, 256, 0, stream>>>(bigB, dis, es, ed, eew, b2, bigA, 16);      // h2

    score_dot<<<cdiv(N2 * 32, 256), 256, 0, stream>>>(bigA, Ws2n, Ws2r, pbuf, rbuf, N2);
    fill_f32<<<cdiv(N2, 256), 256, 0, stream>>>(sbuf, 0.f, N2);
    score_scatter<<<cdiv(E, 256), 256, 0, stream>>>(es, ed, eew, pbuf, sbuf, E);
    score_finish<<<cdiv(N2, 256), 256, 0, stream>>>(sbuf, rbuf, bs2, N2);
    fill_i32<<<cdiv(N2, 256), 256, 0, stream>>>(o2n, -1, N2);
    topk_kernel<<<cdiv(GG, 256), 256, 0, stream>>>(sbuf, topv, perm, o2n, GG, 16, 8);
    gather_gate<<<cdiv(N3 * 128, 256), 256, 0, stream>>>(bigA, perm, topv, bigB, N3); // xk2
    readout_kernel<<<cdiv(GG * 128, 256), 256, 0, stream>>>(bigB, feat, 8, 256, GG);  // x2
    edge_remap<<<cdiv(E, 256), 256, 0, stream>>>(es, ed, eew, es, ed, eew, o2n, 8, E);

    // ================= Layer 3 =================
    gemm(bigB, W3, nullptr, bigA, N3, 128, 128, 0);                   // h_lin3
    fill_f32<<<cdiv(N3, 256), 256, 0, stream>>>(dis, 0.f, N3);
    deg_scatter<<<cdiv(E, 256), 256, 0, stream>>>(ed, eew, dis, E);
    finish_dis<<<cdiv(N3, 256), 256, 0, stream>>>(dis, N3);
    gcn_agg_graph<<<GG, 256, 0, stream>>>(bigA, dis, es, ed, eew, b3, bigB, 8);       // h3

    score_dot<<<cdiv(N3 * 32, 256), 256, 0, stream>>>(bigB, Ws3n, Ws3r, pbuf, rbuf, N3);
    fill_f32<<<cdiv(N3, 256), 256, 0, stream>>>(sbuf, 0.f, N3);
    score_scatter<<<cdiv(E, 256), 256, 0, stream>>>(es, ed, eew, pbuf, sbuf, E);
    score_finish<<<cdiv(N3, 256), 256, 0, stream>>>(sbuf, rbuf, bs3, N3);
    fill_i32<<<cdiv(N3, 256), 256, 0, stream>>>(o2n, -1, N3);
    topk_kernel<<<cdiv(GG, 256), 256, 0, stream>>>(sbuf, topv, perm, o2n, GG, 8, 4);
    gather_gate<<<cdiv(N4 * 128, 256), 256, 0, stream>>>(bigB, perm, topv, bigA, N4); // xk3
    readout_kernel<<<cdiv(GG * 128, 256), 256, 0, stream>>>(bigA, feat, 4, 512, GG);  // x3

    // ================= DDI GCN =================
    gemm(feat, Wd, nullptr, hd, GG, 768, 256, 0);                     // feat @ Wd
    fill_f32<<<cdiv(GG, 256), 256, 0, stream>>>(degd, 0.f, GG);
    ddi_deg<<<cdiv(TWOB, 256), 256, 0, stream>>>(dd, degd, TWOB);
    finish_dis<<<cdiv(GG, 256), 256, 0, stream>>>(degd, GG);
    ddi_init<<<cdiv(GG * 256, 256), 256, 0, stream>>>(hd, degd, zagg, GG * 256);
    ddi_scatter<<<cdiv(TWOB * 256, 256), 256, 0, stream>>>(ds, dd, degd, hd, zagg, TWOB);
    bias_relu_256<<<cdiv(GG * 256, 256), 256, 0, stream>>>(zagg, bd, z, GG * 256);    // z

    // ================= Heads + loss =================
    gather_rows256<<<cdiv(TWOB * 256, 256), 256, 0, stream>>>(z, ds, srcf, TWOB);
    gather_rows256<<<cdiv(TWOB * 256, 256), 256, 0, stream>>>(z, dd, tgtf, TWOB);
    gemm(srcf, Wl1, bl1, Xh, TWOB, 256, 256, 2);    // sigmoid(src @ Wl1 + bl1)
    gemm(tgtf, Wl2, bl2, Yh, TWOB, 256, 256, 2);    // sigmoid(tgt @ Wl2 + bl2)
    gemm(attr, Wl3, bl3, Ah, TWOB, 64, 256, 2);     // sigmoid(attr @ Wl3 + bl3)

    loss_norm<<<cdiv(TWOB * 32, 256), 256, 0, stream>>>(Xh, Yh, Ah, out, B_);
    loss_final<<<cdiv(B_, 256), 256, 0, stream>>>(out, B_);
}